// BestOfKSoftminOT_16088947491323
// MI455X (gfx1250) — compile-verified
//
#include <hip/hip_runtime.h>
#include <math.h>

#define T_ 512
#define KEXP 8
#define B_ 16
#define D_ 32
#define N_ 2048
#define PAD 36
#define INV_EPS 10.0f
#define TAU_ 0.5f
#define ITERS_ 60
#define LOG_A (-6.2383246250395077f) /* -ln(512) */
#define NEG_BIG (-3.4e38f)

typedef __attribute__((ext_vector_type(2))) float v2f;
typedef __attribute__((ext_vector_type(8))) float v8f;

// Load a 16x4 f32 WMMA A/B fragment from LDS (row-major, padded stride).
// lane 0-15: row = base+lane, k = kb+{0,1}; lane 16-31: row = base+lane-16, k = kb+2+{0,1}
__device__ __forceinline__ v2f ldfrag(const float* p, int row_base, int kb, int lane) {
    int r = row_base + (lane & 15);
    int c = kb + ((lane >> 4) << 1);
    return *(const v2f*)(p + r * PAD + c);
}

// Fused GEMM + clamp + online row-logsumexp:
//   out[t] = LOG_A - lse_s( g[s] - max(Pn[t]+Qn[s]-2*(P_t . Q_s), 0) * INV_EPS )
// Each wave handles 4 row-tiles of 16 rows; cross products via v_wmma_f32_16x16x4_f32.
__device__ __forceinline__ void lse_pass(const float* P, const float* Pn,
                                         const float* Q, const float* Qn,
                                         const float* g, float* out,
                                         int lane, int wave) {
    for (int i = 0; i < 4; ++i) {
        const int row_base = ((wave << 2) + i) << 4;
        v2f a[8];
#pragma unroll
        for (int j = 0; j < 8; ++j) a[j] = ldfrag(P, row_base, j * 4, lane);
        const int rofs = row_base + ((lane >> 4) << 3);
        float pn[8], rmax[8], rsum[8];
#pragma unroll
        for (int m = 0; m < 8; ++m) { pn[m] = Pn[rofs + m]; rmax[m] = NEG_BIG; rsum[m] = 0.f; }

        for (int ct = 0; ct < 32; ++ct) {
            const int cb = ct << 4;
            const int n  = cb + (lane & 15);
            const float qn = Qn[n];
            const float gv = g[n];
            v2f bf[8];
#pragma unroll
            for (int j = 0; j < 8; ++j) bf[j] = ldfrag(Q, cb, j * 4, lane);
            v8f acc = {0.f, 0.f, 0.f, 0.f, 0.f, 0.f, 0.f, 0.f};
#pragma unroll
            for (int j = 0; j < 8; ++j)
                acc = __builtin_amdgcn_wmma_f32_16x16x4_f32(
                    false, a[j], false, bf[j], (short)0, acc, false, false);
#pragma unroll
            for (int m = 0; m < 8; ++m) {
                float c   = fmaxf(pn[m] + qn - 2.0f * acc[m], 0.f);
                float val = gv - c * INV_EPS;
                float nm  = fmaxf(rmax[m], val);
                rsum[m]   = rsum[m] * __expf(rmax[m] - nm) + __expf(val - nm);
                rmax[m]   = nm;
            }
        }
        // combine the 16 lanes that share each row (stays within 16-lane halves)
#pragma unroll
        for (int off = 1; off < 16; off <<= 1) {
#pragma unroll
            for (int m = 0; m < 8; ++m) {
                float om = __shfl_xor(rmax[m], off, 32);
                float os = __shfl_xor(rsum[m], off, 32);
                float nm = fmaxf(rmax[m], om);
                rsum[m]  = rsum[m] * __expf(rmax[m] - nm) + os * __expf(om - nm);
                rmax[m]  = nm;
            }
        }
        if ((lane & 15) == 0) {
#pragma unroll
            for (int m = 0; m < 8; ++m)
                out[rofs + m] = LOG_A - (rmax[m] + __logf(rsum[m]));
        }
    }
}

__global__ __launch_bounds__(256) void sinkhorn_ot_kernel(
    const float* __restrict__ sim, const float* __restrict__ expert,
    const int* __restrict__ starts, float* __restrict__ lk_out) {
    extern __shared__ float smem[];
    float* sX  = smem;                 // 512*36
    float* sY  = sX + T_ * PAD;        // 512*36
    float* sXn = sY + T_ * PAD;        // 512
    float* sYn = sXn + T_;             // 512
    float* sU  = sYn + T_;             // 512
    float* sV  = sU + T_;              // 512
    float* sRed = sV + T_;             // 8

    const int bk   = blockIdx.x;
    const int b    = bk >> 3;
    const int kidx = bk & 7;
    const int start = starts[b * KEXP + kidx];
    const int tid  = threadIdx.x;
    const int lane = tid & 31;
    const int wave = tid >> 5;

    // Stage X = sim_seq[b] and Y = expert[b, start:start+T] into LDS (padded rows).
    for (int i = tid; i < T_ * D_; i += 256) {
        int r = i >> 5, c = i & 31;
        sX[r * PAD + c] = sim[(size_t)b * T_ * D_ + i];
        sY[r * PAD + c] = expert[((size_t)b * N_ + start + r) * D_ + c];
    }
    __syncthreads();
    for (int r = tid; r < T_; r += 256) {
        float sx = 0.f, sy = 0.f;
#pragma unroll
        for (int c = 0; c < D_; ++c) {
            float xv = sX[r * PAD + c];
            float yv = sY[r * PAD + c];
            sx += xv * xv; sy += yv * yv;
        }
        sXn[r] = sx; sYn[r] = sy; sV[r] = 0.f;
    }
    __syncthreads();

    // 60 Sinkhorn iterations, fully LDS-resident; C recomputed by WMMA each pass.
    for (int it = 0; it < ITERS_; ++it) {
        lse_pass(sX, sXn, sY, sYn, sV, sU, lane, wave);  // u-update (rows)
        __syncthreads();
        lse_pass(sY, sYn, sX, sXn, sU, sV, lane, wave);  // v-update (cols, transposed GEMM)
        __syncthreads();
    }

    // Lk = sum_{t,s} exp(u[t] + v[s] - C/eps) * C
    float lk = 0.f;
    for (int i = 0; i < 4; ++i) {
        const int row_base = ((wave << 2) + i) << 4;
        v2f a[8];
#pragma unroll
        for (int j = 0; j < 8; ++j) a[j] = ldfrag(sX, row_base, j * 4, lane);
        const int rofs = row_base + ((lane >> 4) << 3);
        float pn[8], uu[8];
#pragma unroll
        for (int m = 0; m < 8; ++m) { pn[m] = sXn[rofs + m]; uu[m] = sU[rofs + m]; }
        for (int ct = 0; ct < 32; ++ct) {
            const int cb = ct << 4;
            const int n  = cb + (lane & 15);
            const float qn = sYn[n];
            const float vv = sV[n];
            v2f bf[8];
#pragma unroll
            for (int j = 0; j < 8; ++j) bf[j] = ldfrag(sY, cb, j * 4, lane);
            v8f acc = {0.f, 0.f, 0.f, 0.f, 0.f, 0.f, 0.f, 0.f};
#pragma unroll
            for (int j = 0; j < 8; ++j)
                acc = __builtin_amdgcn_wmma_f32_16x16x4_f32(
                    false, a[j], false, bf[j], (short)0, acc, false, false);
#pragma unroll
            for (int m = 0; m < 8; ++m) {
                float c = fmaxf(pn[m] + qn - 2.0f * acc[m], 0.f);
                lk += __expf(uu[m] + vv - c * INV_EPS) * c;
            }
        }
    }
#pragma unroll
    for (int off = 16; off > 0; off >>= 1) lk += __shfl_xor(lk, off, 32);
    if (lane == 0) sRed[wave] = lk;
    __syncthreads();
    if (tid == 0) {
        float s = 0.f;
        for (int w = 0; w < 8; ++w) s += sRed[w];
        lk_out[bk] = s;
    }
}

__global__ void finalize_kernel(const float* __restrict__ lk, float* __restrict__ out) {
    if (threadIdx.x == 0 && blockIdx.x == 0) {
        float acc = 0.f;
        for (int b = 0; b < B_; ++b) {
            float z[KEXP];
            float m = NEG_BIG;
            for (int k = 0; k < KEXP; ++k) {
                z[k] = -lk[b * KEXP + k] * (1.0f / TAU_);
                m = fmaxf(m, z[k]);
            }
            float s = 0.f;
            for (int k = 0; k < KEXP; ++k) s += __expf(z[k] - m);
            acc += m + __logf(s);
        }
        out[0] = -TAU_ * acc / (float)B_;
    }
}

extern "C" void kernel_launch(void* const* d_in, const int* in_sizes, int n_in,
                              void* d_out, int out_size, void* d_ws, size_t ws_size,
                              hipStream_t stream) {
    (void)in_sizes; (void)n_in; (void)out_size; (void)ws_size;
    const float* sim    = (const float*)d_in[0];
    const float* expert = (const float*)d_in[1];
    const int*   starts = (const int*)d_in[2];
    float* lk = (float*)d_ws;

    const size_t smem_bytes = (size_t)(2 * T_ * PAD + 4 * T_ + 8) * sizeof(float);
    hipLaunchKernelGGL(sinkhorn_ot_kernel, dim3(B_ * KEXP), dim3(256), smem_bytes, stream,
                       sim, expert, starts, lk);
    hipLaunchKernelGGL(finalize_kernel, dim3(1), dim3(32), 0, stream, lk, (float*)d_out);
}